// MultiQueryAndGroup_54219667144937
// MI455X (gfx1250) — compile-verified
//
#include <hip/hip_runtime.h>
#include <hip/hip_bf16.h>

// ---- problem constants (match reference) ----
constexpr int   Bc = 4;
constexpr int   Nn = 16384;
constexpr int   Ss = 2048;
constexpr int   Cc = 128;
constexpr int   Kk = 32;
constexpr float RAD  = 0.2f;
constexpr float RAD2 = 0.04f;

typedef __attribute__((ext_vector_type(2))) float v2f;
typedef __attribute__((ext_vector_type(8))) float v8f;

// ---------------------------------------------------------------------------
// Kernel T: tiled transpose  src(batch, Rr, Ccols) -> dst(batch, Ccols, Rr)
// Converts (B,C,N) channel-major features to (B,N,C) point-major so the
// gather phase does coalesced 512B row reads instead of 4B scatters.
// ---------------------------------------------------------------------------
__global__ __launch_bounds__(256)
void transpose_kernel(const float* __restrict__ src, float* __restrict__ dst,
                      int Rr, int Ccols) {
  __shared__ float tile[32][33];
  const int c0 = blockIdx.x * 32;
  const int r0 = blockIdx.y * 32;
  const float* s = src + (size_t)blockIdx.z * Rr * Ccols;
  float*       d = dst + (size_t)blockIdx.z * Rr * Ccols;
  for (int i = threadIdx.y; i < 32; i += 8) {
    tile[i][threadIdx.x] = s[(size_t)(r0 + i) * Ccols + (c0 + threadIdx.x)];
  }
  __syncthreads();
  for (int i = threadIdx.y; i < 32; i += 8) {
    d[(size_t)(c0 + i) * Rr + (r0 + threadIdx.x)] = tile[threadIdx.x][i];
  }
}

// ---------------------------------------------------------------------------
// Kernel 1: fused distance (WMMA f32 16x16x4) + exact top-K selection.
// Block = 64 threads = 2 waves; each wave owns 16 centers.
// d2 tile:  A row m = [-2cx, -2cy, -2cz, 1]  (K=4)
//           B col n = [ px,   py,   pz,  |p|^2]
//           C       = |c|^2  broadcast per row
//  => D[m][n] = |c|^2 + |p|^2 - 2 c.p  exactly, one WMMA per 16x16 tile.
// Point chunks are staged with GLOBAL_LOAD_ASYNC_TO_LDS_B128 (ASYNCcnt).
// ---------------------------------------------------------------------------
__global__ __launch_bounds__(64)
void topk_kernel(const float* __restrict__ points_xyz,
                 const float* __restrict__ center_xyz,
                 float* __restrict__ out_d, int* __restrict__ out_i) {
  __shared__ float  ptsraw[3072];     // 1024 points * xyz, raw stream (12 KB)
  __shared__ float  bestd[32 * 32];   // 32 centers x 32 kept dists
  __shared__ int    besti[32 * 32];
  __shared__ float  curmax[32];       // current K-th distance per center
  __shared__ float  cn_sh[32];        // |c|^2 per center

  const int tid  = threadIdx.x;
  const int wave = tid >> 5;
  const int lane = tid & 31;
  const int hi   = lane >> 4;   // 0: lanes 0-15, 1: lanes 16-31
  const int l16  = lane & 15;

  const int tiles_per_b = Ss / 32;
  const int b     = blockIdx.x / tiles_per_b;
  const int ctile = blockIdx.x % tiles_per_b;
  const int s0    = ctile * 32 + wave * 16;   // first center of this wave

  for (int i = tid; i < 32 * 32; i += 64) { bestd[i] = 3.0e38f; besti[i] = 0; }
  if (tid < 32) curmax[tid] = 3.0e38f;

  // Center coords for center (s0 + l16); both lane halves read the same row.
  const float* cb = center_xyz + ((size_t)b * Ss + (s0 + l16)) * 3;
  const float cx = cb[0], cy = cb[1], cz = cb[2];
  const float cn = cx * cx + cy * cy + cz * cz;
  if (hi == 0) cn_sh[wave * 16 + l16] = cn;

  // A fragment (16x4 f32): lanes0-15 hold K=0,1; lanes16-31 hold K=2,3.
  v2f amat;
  amat.x = (hi == 0) ? -2.0f * cx : -2.0f * cz;
  amat.y = (hi == 0) ? -2.0f * cy :  1.0f;

  __syncthreads();

  v8f cacc;
#pragma unroll
  for (int r = 0; r < 8; ++r) cacc[r] = cn_sh[wave * 16 + r + 8 * hi];

  const unsigned ldsbase = (unsigned)(uintptr_t)(&ptsraw[0]);

  for (int base = 0; base < Nn; base += 1024) {
    // Async-stage 1024 points (12288 B = 768 x B128) straight into LDS.
    const float* gsrc = points_xyz + ((size_t)b * Nn + base) * 3;
    for (int j = tid; j < 768; j += 64) {
      const unsigned long long ga =
          (unsigned long long)(uintptr_t)gsrc + (unsigned long long)j * 16u;
      const unsigned la = ldsbase + (unsigned)j * 16u;
      asm volatile("global_load_async_to_lds_b128 %0, %1, off"
                   :: "v"(la), "v"(ga) : "memory");
    }
    if (base + 1024 < Nn)
      __builtin_prefetch(points_xyz + ((size_t)b * Nn + base + 1024) * 3, 0, 1);
    asm volatile("s_wait_asynccnt 0x0" ::: "memory");
    __syncthreads();

    for (int sub = 0; sub < 64; ++sub) {
      const int nloc = sub * 16 + l16;
      const float px = ptsraw[nloc * 3 + 0];
      const float py = ptsraw[nloc * 3 + 1];
      const float pz = ptsraw[nloc * 3 + 2];
      const float pn = px * px + py * py + pz * pz;
      v2f bmat;
      bmat.x = (hi == 0) ? px : pz;
      bmat.y = (hi == 0) ? py : pn;

      // D = A*B + C : 16x16 tile of squared distances.
      v8f d8 = __builtin_amdgcn_wmma_f32_16x16x4_f32(
          false, amat, false, bmat, (short)0, cacc, false, false);

      const int n = base + nloc;
#pragma unroll
      for (int r = 0; r < 8; ++r) {
        const int c = wave * 16 + r + 8 * hi;        // shared-array center slot
        const float d = fmaxf(d8[r], 0.0f);
        const bool cand = d < curmax[c];
        unsigned m = (unsigned)__ballot(cand);
        while (m) {                                   // uniform loop
          const int j = __ffs(m) - 1;
          m &= m - 1;
          // Uniform-lane broadcasts: v_readlane, not ds_bpermute.
          const float dj =
              __int_as_float(__builtin_amdgcn_readlane(__float_as_int(d), j));
          const int   nj = __builtin_amdgcn_readlane(n, j);
          const int   cj = __builtin_amdgcn_readlane(c, j);
          // Wave-parallel replace-max on list cj: lane owns slot 'lane'.
          float slotd = bestd[cj * 32 + lane];
          float mval = slotd; int mslot = lane;
#pragma unroll
          for (int off = 16; off > 0; off >>= 1) {
            const float ov = __shfl_xor(mval, off);
            const int   os = __shfl_xor(mslot, off);
            if (ov > mval || (ov == mval && os < mslot)) { mval = ov; mslot = os; }
          }
          if (dj < mval) {                            // uniform branch
            if (lane == mslot) {
              bestd[cj * 32 + lane] = dj;
              besti[cj * 32 + lane] = nj;
              slotd = dj;
            }
            float nm = slotd;
#pragma unroll
            for (int off = 16; off > 0; off >>= 1)
              nm = fmaxf(nm, __shfl_xor(nm, off));
            if (lane == 0) curmax[cj] = nm;
          }
        }
      }
    }
    __syncthreads();
  }

  // Bitonic sort each of this wave's 16 lists ascending, then emit.
  for (int cc = 0; cc < 16; ++cc) {
    const int c = wave * 16 + cc;
    float d = bestd[c * 32 + lane];
    int   i = besti[c * 32 + lane];
#pragma unroll
    for (int kk = 2; kk <= 32; kk <<= 1) {
#pragma unroll
      for (int jj = kk >> 1; jj > 0; jj >>= 1) {
        const int partner = lane ^ jj;
        const float od = __shfl_xor(d, jj);
        const int   oi = __shfl_xor(i, jj);
        const bool up      = ((lane & kk) == 0);
        const bool takeMin = (lane < partner) == up;
        const bool pick    = takeMin ? (od < d) : (od > d);
        if (pick) { d = od; i = oi; }
      }
    }
    const size_t o = ((size_t)b * Ss + (s0 + cc)) * Kk + lane;
    out_d[o] = d;
    out_i[o] = i;
  }
}

// ---------------------------------------------------------------------------
// Kernel G: one wave per (b,s,k). Coalesced float4 gather from the transposed
// feature tensors, radius mask, relative-xyz normalize, 131-float output row.
// ---------------------------------------------------------------------------
__global__ __launch_bounds__(128)
void gather_kernel(const float* __restrict__ points_xyz,
                   const float* __restrict__ center_xyz,
                   const float* __restrict__ feaT,   // (B,N,C)
                   const float* __restrict__ cenT,   // (B,S,C)
                   const float* __restrict__ tk_d,
                   const int*   __restrict__ tk_i,
                   float* __restrict__ out) {
  const int lane = threadIdx.x & 31;
  const int wv   = threadIdx.x >> 5;
  const long long g = (long long)blockIdx.x * 4 + wv;   // wave id over B*S*K
  const int k = (int)(g % Kk);
  const long long bs = g / Kk;
  const int s = (int)(bs % Ss);
  const int b = (int)(bs / Ss);

  const size_t tko = ((size_t)b * Ss + s) * Kk + k;
  const int   idx  = tk_i[tko];
  const float d    = tk_d[tko];
  const bool  mask = (k > 0) && (d > RAD2);

  float* orow = out + tko * (size_t)(3 + Cc);

  if (lane < 3) {
    const float pv = points_xyz[((size_t)b * Nn + idx) * 3 + lane];
    const float cv = center_xyz[((size_t)b * Ss + s)   * 3 + lane];
    orow[lane] = mask ? 0.0f : (pv - cv) * (1.0f / RAD);
  }

  const float4* src = mask
      ? (const float4*)(cenT + ((size_t)b * Ss + s)   * Cc)
      : (const float4*)(feaT + ((size_t)b * Nn + idx) * Cc);
  const float4 f = src[lane];
  float* fo = orow + 3 + lane * 4;
  fo[0] = f.x; fo[1] = f.y; fo[2] = f.z; fo[3] = f.w;
}

// ---------------------------------------------------------------------------
extern "C" void kernel_launch(void* const* d_in, const int* in_sizes, int n_in,
                              void* d_out, int out_size, void* d_ws, size_t ws_size,
                              hipStream_t stream) {
  const float* points_xyz = (const float*)d_in[0];   // (B,N,3)
  const float* center_xyz = (const float*)d_in[1];   // (B,S,3)
  const float* points_fea = (const float*)d_in[2];   // (B,C,N)
  const float* center_fea = (const float*)d_in[3];   // (B,C,S)

  float* ws   = (float*)d_ws;
  float* feaT = ws;                                   // B*N*C floats (33.5 MB)
  float* cenT = feaT + (size_t)Bc * Nn * Cc;          // B*S*C floats ( 4.0 MB)
  float* tkd  = cenT + (size_t)Bc * Ss * Cc;          // B*S*K floats ( 1.0 MB)
  int*   tki  = (int*)(tkd + (size_t)Bc * Ss * Kk);   // B*S*K ints   ( 1.0 MB)

  dim3 blkT(32, 8, 1);
  transpose_kernel<<<dim3(Nn / 32, Cc / 32, Bc), blkT, 0, stream>>>(
      points_fea, feaT, Cc, Nn);
  transpose_kernel<<<dim3(Ss / 32, Cc / 32, Bc), blkT, 0, stream>>>(
      center_fea, cenT, Cc, Ss);

  topk_kernel<<<Bc * (Ss / 32), 64, 0, stream>>>(points_xyz, center_xyz, tkd, tki);

  gather_kernel<<<(Bc * Ss * Kk) / 4, 128, 0, stream>>>(
      points_xyz, center_xyz, feaT, cenT, tkd, tki, (float*)d_out);
}